// GroupedLinear_50629074486047
// MI455X (gfx1250) — compile-verified
//
#include <hip/hip_runtime.h>

// GroupedLinear on gfx1250 (MI455X): 8 x [8192x512x512] f32 GEMM with fused
// channel gather, using V_WMMA_F32_16X16X4_F32 (exact f32 matrix path).
//
// v2: M/N-major LDS tiles with contiguous k-pairs -> single ds_load_b64 per
// WMMA fragment (no repack movs), b128 LDS fills, and a register-staged
// pipeline overlapping the next tile's global gather with the WMMA burst.

typedef __attribute__((ext_vector_type(2))) float v2f;
typedef __attribute__((ext_vector_type(4))) float v4f;
typedef __attribute__((ext_vector_type(8))) float v8f;
typedef __attribute__((ext_vector_type(4))) int   v4i;

namespace {
constexpr int BATCH = 8192;
constexpr int CIN   = 4096;
constexpr int G     = 8;
constexpr int CG    = 512;   // channels per group (K)
constexpr int J     = 512;   // outputs per group (N per group)
constexpr int OUTW  = G * J; // 4096

constexpr int BM = 128;      // block tile M
constexpr int BN = 128;      // block tile N
constexpr int BK = 32;       // block tile K
// Row stride 36 dwords = 144 B: multiple of 16 B (b128 stores, b64 frags ok)
// and 36*col mod 64 distinct for col 0..15 -> conflict-free fragment reads.
constexpr int LDK = BK + 4;
} // namespace

__global__ __launch_bounds__(256) void grouped_linear_wmma_f32(
    const float* __restrict__ x,      // [BATCH, CIN]
    const int*   __restrict__ arr,    // [G, CG] channel permutation
    const float* __restrict__ w,      // [G, J, CG]
    const float* __restrict__ bias,   // [G, J]
    float*       __restrict__ out)    // [BATCH, OUTW]
{
    __shared__ float sA[BM][LDK];     // sA[m][k]: k-pairs contiguous
    __shared__ float sW[BN][LDK];     // sW[n][k]: k-pairs contiguous
    __shared__ int   sIdx[CG];        // this group's gather indices

    const int g    = blockIdx.z;
    const int m0   = blockIdx.y * BM;
    const int n0   = blockIdx.x * BN;
    const int tid  = threadIdx.x;
    const int lane = tid & 31;        // wave32
    const int wave = tid >> 5;        // 8 waves
    const int wm   = wave & 3;        // 4 waves tile M: 32 rows each
    const int wn   = wave >> 2;       // 2 waves tile N: 64 cols each
    const int lhalf = lane >> 4;      // 0 | 1 (lane half)
    const int l16   = lane & 15;

    // Cache the group's permutation once (512 ints = 2 KB).
    for (int i = tid; i < CG; i += 256) sIdx[i] = arr[g * CG + i];
    __syncthreads();

    // Per-thread fill mapping (identical for A and W tiles):
    // 2 threads per row; each covers 16 consecutive k.
    const int e   = tid * 16;
    const int row = e >> 5;           // 0..127
    const int kk  = e & 31;           // 0 or 16
    const float* xrow = x + (size_t)(m0 + row) * CIN;
    const float* wrow = w + (size_t)g * J * CG + (size_t)(n0 + row) * CG;

    v4f ra[4];                        // staged A (gathered) : 16 floats
    v4f rw[4];                        // staged W            : 16 floats

    auto preload = [&](int k0) {
        v4i idx[4];
        #pragma unroll
        for (int q4 = 0; q4 < 4; ++q4)
            idx[q4] = *(const v4i*)&sIdx[k0 + kk + 4 * q4];   // ds_load_b128
        #pragma unroll
        for (int q4 = 0; q4 < 4; ++q4)
            #pragma unroll
            for (int q = 0; q < 4; ++q)
                ra[q4][q] = xrow[idx[q4][q]];                 // 4B L2 gathers
        #pragma unroll
        for (int q4 = 0; q4 < 4; ++q4)
            rw[q4] = *(const v4f*)(wrow + k0 + kk + 4 * q4);  // global b128
    };

    auto stage_to_lds = [&]() {
        #pragma unroll
        for (int q4 = 0; q4 < 4; ++q4)
            *(v4f*)&sA[row][kk + 4 * q4] = ra[q4];            // ds_store_b128
        #pragma unroll
        for (int q4 = 0; q4 < 4; ++q4)
            *(v4f*)&sW[row][kk + 4 * q4] = rw[q4];            // ds_store_b128
    };

    v8f acc[2][4];
    #pragma unroll
    for (int i = 0; i < 2; ++i)
        #pragma unroll
        for (int j = 0; j < 4; ++j)
            acc[i][j] = v8f{0.f, 0.f, 0.f, 0.f, 0.f, 0.f, 0.f, 0.f};

    preload(0);

    for (int k0 = 0; k0 < CG; k0 += BK) {
        stage_to_lds();
        __syncthreads();

        if (k0 + BK < CG) preload(k0 + BK);   // overlap with WMMA burst

        // 8 k-steps of 4 -> 64 v_wmma_f32_16x16x4_f32 per tile.
        #pragma unroll
        for (int ks = 0; ks < BK; ks += 4) {
            // ISA 7.12.2: A 16x4 -> lanes 0-15 hold K{0,1}, lanes 16-31 K{2,3};
            // krow is even, so each fragment is one aligned 8B LDS read.
            const int krow = ks + (lhalf << 1);
            v2f aF[2], bF[4];
            #pragma unroll
            for (int i = 0; i < 2; ++i)
                aF[i] = *(const v2f*)&sA[wm * 32 + i * 16 + l16][krow];
            #pragma unroll
            for (int j = 0; j < 4; ++j)
                bF[j] = *(const v2f*)&sW[wn * 64 + j * 16 + l16][krow];
            #pragma unroll
            for (int i = 0; i < 2; ++i)
                #pragma unroll
                for (int j = 0; j < 4; ++j)
                    acc[i][j] = __builtin_amdgcn_wmma_f32_16x16x4_f32(
                        /*neg_a=*/false, aF[i], /*neg_b=*/false, bF[j],
                        /*c_mod=*/(short)0, acc[i][j],
                        /*reuse_a=*/false, /*reuse_b=*/false);
        }
        __syncthreads();   // LDS reads done before next stage_to_lds()
    }

    // ---- epilogue: bias + store (C layout: VGPR r -> M=r + 8*lhalf, N=l16) ----
    const float* bg = bias + g * J;
    #pragma unroll
    for (int i = 0; i < 2; ++i) {
        const int rowBase = m0 + wm * 32 + i * 16 + (lhalf << 3);
        #pragma unroll
        for (int j = 0; j < 4; ++j) {
            const int col = wn * 64 + j * 16 + l16;
            const float bv = bg[n0 + col];
            float* o = out + (size_t)rowBase * OUTW + g * J + n0 + col;
            #pragma unroll
            for (int r = 0; r < 8; ++r)
                o[(size_t)r * OUTW] = acc[i][j][r] + bv;
        }
    }
}

extern "C" void kernel_launch(void* const* d_in, const int* in_sizes, int n_in,
                              void* d_out, int out_size, void* d_ws, size_t ws_size,
                              hipStream_t stream) {
    (void)in_sizes; (void)n_in; (void)out_size; (void)d_ws; (void)ws_size;
    const float* x    = (const float*)d_in[0];   // [8192, 4096] f32
    const int*   arr  = (const int*)d_in[1];     // [8, 512] int32 (JAX default int)
    const float* w    = (const float*)d_in[2];   // [8, 512, 512] f32
    const float* bias = (const float*)d_in[3];   // [8, 512] f32
    float*       out  = (float*)d_out;           // [8192, 4096] f32

    dim3 grid(J / BN, BATCH / BM, G);            // (4, 64, 8)
    dim3 block(256);                             // 8 wave32 per workgroup
    grouped_linear_wmma_f32<<<grid, block, 0, stream>>>(x, arr, w, bias, out);
}